// RecLGN_66383014527389
// MI455X (gfx1250) — compile-verified
//
#include <hip/hip_runtime.h>

namespace {
constexpr int NU   = 100000;
constexpr int NR   = 200000;
constexpr int D    = 128;
constexpr int NE   = 1500000;
constexpr float ALPHA = 0.25f;   // 1/(NUM_LAYERS+1)
}

typedef float v2f __attribute__((ext_vector_type(2)));
typedef float v8f __attribute__((ext_vector_type(8)));

// ---------------------------------------------------------------- zero fill
__global__ void k_zero(float4* __restrict__ p, int n4) {
    int i = blockIdx.x * blockDim.x + threadIdx.x;
    if (i < n4) p[i] = make_float4(0.f, 0.f, 0.f, 0.f);
}

// ------------------------------------------------------- weighted degrees
__global__ void k_deg(const float* __restrict__ w, const int* __restrict__ ui,
                      const int* __restrict__ ri, float* __restrict__ du,
                      float* __restrict__ dr) {
    int e = blockIdx.x * blockDim.x + threadIdx.x;
    if (e >= NE) return;
    float we = w[e];
    unsafeAtomicAdd(&du[ui[e]], we);   // hardware global_atomic_add_f32
    unsafeAtomicAdd(&dr[ri[e]], we);
}

// ------------------------------------------------------------ edge norms
__global__ void k_norm(const float* __restrict__ w, const int* __restrict__ ui,
                       const int* __restrict__ ri, const float* __restrict__ du,
                       const float* __restrict__ dr, float* __restrict__ nrm) {
    int e = blockIdx.x * blockDim.x + threadIdx.x;
    if (e >= NE) return;
    float p = du[ui[e]] * dr[ri[e]];
    nrm[e] = w[e] * rsqrtf(fmaxf(p, 1e-12f));
}

// ------------------------------------------- init: usr buffer + usr_out
__global__ void k_init_usr(const float4* __restrict__ emb, float4* __restrict__ buf,
                           float4* __restrict__ out, int n4) {
    int i = blockIdx.x * blockDim.x + threadIdx.x;
    if (i >= n4) return;
    float4 v = emb[i];
    buf[i] = v;
    out[i] = make_float4(ALPHA * v.x, ALPHA * v.y, ALPHA * v.z, ALPHA * v.w);
}

// ------------------- init: rec_out = alpha * concat(rcp_emb, recipe_feat)
__global__ void k_init_rec(const float4* __restrict__ emb, const float4* __restrict__ feat,
                           float4* __restrict__ out) {
    int i = blockIdx.x * blockDim.x + threadIdx.x;      // over NR*32 float4s
    if (i >= NR * 32) return;
    int row = i >> 5, q = i & 31;                       // 32 float4s per 128-wide row
    float4 v = (q < 8) ? emb[row * 8 + q]               // EMB=32 -> 8 float4s
                       : feat[row * 24 + (q - 8)];      // RDIM=96 -> 24 float4s
    out[(size_t)row * 32 + q] =
        make_float4(ALPHA * v.x, ALPHA * v.y, ALPHA * v.z, ALPHA * v.w);
}

// --------------------------- gather-scale-scatter: one wave32 per edge
// dst[didx[e], :] += src[sidx[e], :] * nrm[e]   (128 floats, 4/lane)
// Edge id is forced scalar so index/norm reads lower to s_load (SMEM) and the
// row-base address math stays on the SALU, leaving VMEM for gather + atomics.
__global__ void k_scatter(const float* __restrict__ src, float* __restrict__ dst,
                          const int* __restrict__ sidx, const int* __restrict__ didx,
                          const float* __restrict__ nrm) {
    int gid  = blockIdx.x * blockDim.x + threadIdx.x;
    int e    = __builtin_amdgcn_readfirstlane(gid >> 5);   // provably wave-uniform
    int lane = threadIdx.x & 31;
    if (e >= NE) return;
    int   s  = sidx[e];                                    // scalar loads
    int   d  = didx[e];
    float nv = nrm[e];
    const float* sp = src + (size_t)s * D;                 // uniform row base
    float*       dp = dst + (size_t)d * D + lane * 4;
    float4 v = ((const float4*)sp)[lane];                  // coalesced 256B/16-lane
    unsafeAtomicAdd(dp + 0, v.x * nv);
    unsafeAtomicAdd(dp + 1, v.y * nv);
    unsafeAtomicAdd(dp + 2, v.z * nv);
    unsafeAtomicAdd(dp + 3, v.w * nv);
}

// ------------------------------- out += alpha * x  via V_WMMA_F32_16X16X4_F32
// One wave per 16x16 tile; block of 256 covers 16 rows x 128 cols.
// D_tile = sum_j A_j * B_j + C_tile, A_j = X[:,4j:4j+4], B_j = alpha-shifted identity.
__global__ void k_axpy_wmma(const float* __restrict__ x, float* __restrict__ out) {
    int tileRow = blockIdx.x;            // rows [16*tileRow, 16*tileRow+16)
    int wave    = threadIdx.x >> 5;      // 0..7 -> 16-col tile
    int lane    = threadIdx.x & 31;
    int n       = lane & 15;             // col within tile (C/D + A row M)
    int half    = lane >> 4;             // 0/1

    const float* xs = x   + (size_t)tileRow * 16 * D + wave * 16;
    float*       os = out + (size_t)tileRow * 16 * D + wave * 16;

    // C layout: vgpr v, lanes 0-15 -> row v, lanes 16-31 -> row v+8
    v8f c;
#pragma unroll
    for (int v = 0; v < 8; ++v) c[v] = os[(v + 8 * half) * D + n];

#pragma unroll
    for (int j = 0; j < 4; ++j) {
        // A 16x4 layout: lanes 0-15 M=lane hold K=0,1; lanes 16-31 hold K=2,3
        v2f a;
        a[0] = xs[n * D + 4 * j + 2 * half + 0];
        a[1] = xs[n * D + 4 * j + 2 * half + 1];
        // B 4x16: row k striped across lanes; B_j[k][col] = alpha*(col == 4j+k)
        v2f b;
        b[0] = (n == 4 * j + 2 * half + 0) ? ALPHA : 0.0f;
        b[1] = (n == 4 * j + 2 * half + 1) ? ALPHA : 0.0f;
        c = __builtin_amdgcn_wmma_f32_16x16x4_f32(false, a, false, b,
                                                  (short)0, c, false, false);
    }

#pragma unroll
    for (int v = 0; v < 8; ++v) os[(v + 8 * half) * D + n] = c[v];
}

// ---------------------------------------------------------------- driver
extern "C" void kernel_launch(void* const* d_in, const int* in_sizes, int n_in,
                              void* d_out, int out_size, void* d_ws, size_t ws_size,
                              hipStream_t stream) {
    const float* usr_emb = (const float*)d_in[0];   // [NU,128]
    const float* rcp_emb = (const float*)d_in[1];   // [NR,32]
    const float* feat    = (const float*)d_in[2];   // [NR,96]
    const float* ew      = (const float*)d_in[3];   // [NE]
    const int*   ui      = (const int*)d_in[4];     // [NE]
    const int*   ri      = (const int*)d_in[5];     // [NE]

    float* usr_out = (float*)d_out;                  // [NU,128]
    float* rec_out = usr_out + (size_t)NU * D;       // [NR,128]

    float* ws   = (float*)d_ws;
    float* du   = ws;                ws += NU;
    float* dr   = ws;                ws += NR;            // contiguous after du
    float* nrm  = ws;                ws += NE;
    float* ubuf = ws;                ws += (size_t)NU * D;
    float* rbuf = ws;                                      // [NR,128]

    auto cdiv = [](long a, long b) { return (int)((a + b - 1) / b); };
    const int T = 256;

    // degrees + norms
    k_zero<<<cdiv((NU + NR) / 4, T), T, 0, stream>>>((float4*)du, (NU + NR) / 4);
    k_deg <<<cdiv(NE, T), T, 0, stream>>>(ew, ui, ri, du, dr);
    k_norm<<<cdiv(NE, T), T, 0, stream>>>(ew, ui, ri, du, dr, nrm);

    // init node state and alpha-scaled outputs (fully overwrites poisoned d_out)
    k_init_usr<<<cdiv((size_t)NU * D / 4, T), T, 0, stream>>>(
        (const float4*)usr_emb, (float4*)ubuf, (float4*)usr_out, NU * D / 4);
    k_init_rec<<<cdiv((size_t)NR * 32, T), T, 0, stream>>>(
        (const float4*)rcp_emb, (const float4*)feat, (float4*)rec_out);

    const int edgeBlocks = cdiv((size_t)NE * 32, T);   // one wave32 per edge
    for (int l = 0; l < 3; ++l) {
        // user -> recipe
        k_zero<<<cdiv((size_t)NR * D / 4, T), T, 0, stream>>>((float4*)rbuf, NR * D / 4);
        k_scatter<<<edgeBlocks, T, 0, stream>>>(ubuf, rbuf, ui, ri, nrm);
        k_axpy_wmma<<<NR / 16, T, 0, stream>>>(rbuf, rec_out);
        // recipe -> user
        k_zero<<<cdiv((size_t)NU * D / 4, T), T, 0, stream>>>((float4*)ubuf, NU * D / 4);
        k_scatter<<<edgeBlocks, T, 0, stream>>>(rbuf, ubuf, ri, ui, nrm);
        k_axpy_wmma<<<NU / 16, T, 0, stream>>>(ubuf, usr_out);
    }
}